// BCE2D_8770323218588
// MI455X (gfx1250) — compile-verified
//
#include <hip/hip_runtime.h>

// Weighted (class-balanced) BCE over N=32,C=1,H=1024,W=1024 f32 tensors.
// Memory-bound streaming reduction: floor ~= 268 MB / 23.3 TB/s ~= 11.5 us.
//
// Single pass, log2-domain accumulation (ln2 scaling deferred to finalize):
//   per element: l = max(log2(t ? p : 1-p), -100/ln2)
//   accumulate:  s_tl = sum t*l,  s_l = sum l,  cnt = sum t
// Then S_pos = ln2*s_tl, S_neg = ln2*(s_l - s_tl), pos = cnt, neg = N - pos,
//   loss = -((neg/N)*S_pos + (pos/N)*S_neg) / N
// Hot loop is 6 VALU + 1 v_log_f32 per element -> stays under the HBM floor.

#define NBLOCKS    1024
#define NTHREADS   256          // 8 wave32 per block on gfx1250
#define WAVESZ     32

// -100 / ln(2): clamp applied in log2 domain so no per-element ln2 multiply.
#define LOG2_CLAMP (-144.26950408889634f)

__global__ __launch_bounds__(NTHREADS)
void bce_partials_kernel(const float4* __restrict__ p4,
                         const float4* __restrict__ t4,
                         float* __restrict__ partials,   // [3 * NBLOCKS]
                         int n4, int n_tail_start, int n_total)
{
    float s_tl = 0.0f;   // sum of t * clamp(log2 q)   (pos-class log sum, /ln2)
    float s_l  = 0.0f;   // sum of     clamp(log2 q)   (all elements,      /ln2)
    float cnt  = 0.0f;   // count of t==1 (exact in f32: <= 32768 per block)

    const int tid    = blockIdx.x * NTHREADS + threadIdx.x;
    const int stride = NBLOCKS * NTHREADS;

    #pragma unroll 4
    for (int i = tid; i < n4; i += stride) {
        const float4 pv = p4[i];          // global_load_b128
        const float4 tv = t4[i];          // global_load_b128

        {   float tt = tv.x, pp = pv.x;
            float q  = (tt > 0.5f) ? pp : (1.0f - pp);
            float l  = fmaxf(__log2f(q), LOG2_CLAMP);
            s_tl = fmaf(tt, l, s_tl);
            s_l += l;
            cnt += tt; }
        {   float tt = tv.y, pp = pv.y;
            float q  = (tt > 0.5f) ? pp : (1.0f - pp);
            float l  = fmaxf(__log2f(q), LOG2_CLAMP);
            s_tl = fmaf(tt, l, s_tl);
            s_l += l;
            cnt += tt; }
        {   float tt = tv.z, pp = pv.z;
            float q  = (tt > 0.5f) ? pp : (1.0f - pp);
            float l  = fmaxf(__log2f(q), LOG2_CLAMP);
            s_tl = fmaf(tt, l, s_tl);
            s_l += l;
            cnt += tt; }
        {   float tt = tv.w, pp = pv.w;
            float q  = (tt > 0.5f) ? pp : (1.0f - pp);
            float l  = fmaxf(__log2f(q), LOG2_CLAMP);
            s_tl = fmaf(tt, l, s_tl);
            s_l += l;
            cnt += tt; }
    }

    // Scalar tail (n not divisible by 4) handled by global thread 0 only.
    if (tid == 0 && n_tail_start < n_total) {
        const float* ps = (const float*)p4;
        const float* ts = (const float*)t4;
        for (int i = n_tail_start; i < n_total; ++i) {
            float tt = ts[i], pp = ps[i];
            float q  = (tt > 0.5f) ? pp : (1.0f - pp);
            float l  = fmaxf(__log2f(q), LOG2_CLAMP);
            s_tl = fmaf(tt, l, s_tl);
            s_l += l;
            cnt += tt;
        }
    }

    // ---- wave32 butterfly reduction (gfx1250 is wave32-only) ----
    #pragma unroll
    for (int off = WAVESZ / 2; off > 0; off >>= 1) {
        s_tl += __shfl_xor(s_tl, off, WAVESZ);
        s_l  += __shfl_xor(s_l,  off, WAVESZ);
        cnt  += __shfl_xor(cnt,  off, WAVESZ);
    }

    // ---- cross-wave reduction through LDS (8 waves / block) ----
    __shared__ float sm[3][NTHREADS / WAVESZ];
    const int wave = threadIdx.x >> 5;
    const int lane = threadIdx.x & (WAVESZ - 1);
    if (lane == 0) {
        sm[0][wave] = s_tl;
        sm[1][wave] = s_l;
        sm[2][wave] = cnt;
    }
    __syncthreads();
    if (threadIdx.x == 0) {
        float a = 0.0f, b = 0.0f, c = 0.0f;
        #pragma unroll
        for (int w = 0; w < NTHREADS / WAVESZ; ++w) {
            a += sm[0][w]; b += sm[1][w]; c += sm[2][w];
        }
        partials[0 * NBLOCKS + blockIdx.x] = a;   // sum t*l  partial
        partials[1 * NBLOCKS + blockIdx.x] = b;   // sum l    partial
        partials[2 * NBLOCKS + blockIdx.x] = c;   // pos count partial
    }
}

// Deterministic finalize: one block combines the NBLOCKS partials in double
// (pos count can reach 2^25 > 2^24, so f32 accumulation would lose exactness),
// applies the deferred ln2 scale, and writes the scalar loss.
__global__ __launch_bounds__(NTHREADS)
void bce_finalize_kernel(const float* __restrict__ partials,
                         float* __restrict__ out,
                         float n_total)
{
    double a = 0.0, b = 0.0, c = 0.0;
    for (int i = threadIdx.x; i < NBLOCKS; i += NTHREADS) {
        a += (double)partials[0 * NBLOCKS + i];
        b += (double)partials[1 * NBLOCKS + i];
        c += (double)partials[2 * NBLOCKS + i];
    }
    #pragma unroll
    for (int off = WAVESZ / 2; off > 0; off >>= 1) {
        a += __shfl_xor(a, off, WAVESZ);
        b += __shfl_xor(b, off, WAVESZ);
        c += __shfl_xor(c, off, WAVESZ);
    }
    __shared__ double sm[3][NTHREADS / WAVESZ];
    const int wave = threadIdx.x >> 5;
    const int lane = threadIdx.x & (WAVESZ - 1);
    if (lane == 0) { sm[0][wave] = a; sm[1][wave] = b; sm[2][wave] = c; }
    __syncthreads();
    if (threadIdx.x == 0) {
        double A = 0.0, L = 0.0, C = 0.0;
        #pragma unroll
        for (int w = 0; w < NTHREADS / WAVESZ; ++w) {
            A += sm[0][w]; L += sm[1][w]; C += sm[2][w];
        }
        const double LN2     = 0.6931471805599453094172321214581766;
        const double total   = (double)n_total;
        const double pos_num = C;
        const double neg_num = total - pos_num;
        const double S_pos   = LN2 * A;          // sum_{t=1} clamp(ln p)
        const double S_neg   = LN2 * (L - A);    // sum_{t=0} clamp(ln(1-p))
        // loss = -mean( w * (t*logp + (1-t)*log1mp) )
        //      = -((neg/total)*S_pos + (pos/total)*S_neg) / total
        const double loss = -((neg_num / total) * S_pos +
                              (pos_num / total) * S_neg) / total;
        out[0] = (float)loss;
    }
}

extern "C" void kernel_launch(void* const* d_in, const int* in_sizes, int n_in,
                              void* d_out, int out_size, void* d_ws, size_t ws_size,
                              hipStream_t stream)
{
    const float4* p4 = (const float4*)d_in[0];   // input probabilities, f32
    const float4* t4 = (const float4*)d_in[1];   // targets in {0,1},   f32
    float* out       = (float*)d_out;
    float* partials  = (float*)d_ws;             // 3 * NBLOCKS floats = 12 KB

    const int n  = in_sizes[0];                  // 32*1*1024*1024 = 33,554,432
    const int n4 = n >> 2;                       // float4 count
    const int tail_start = n4 << 2;

    bce_partials_kernel<<<NBLOCKS, NTHREADS, 0, stream>>>(
        p4, t4, partials, n4, tail_start, n);

    bce_finalize_kernel<<<1, NTHREADS, 0, stream>>>(
        partials, out, (float)n);
}